// History_76836964926238
// MI455X (gfx1250) — compile-verified
//
#include <hip/hip_runtime.h>

// ---------------------------------------------------------------------------
// MI455X (gfx1250) fused History/GRU kernel.
// fp32 WMMA (V_WMMA_F32_16X16X4_F32) for all four GEMMs; workload is
// HBM-bound on the 1.5GB history copy, so fp32 costs nothing vs bf16.
// ---------------------------------------------------------------------------

typedef __attribute__((ext_vector_type(2))) float v2f;
typedef __attribute__((ext_vector_type(8))) float v8f;

#define DD    128
#define LDSP  (DD + 2)   // pad LDS stride (130) -> rows hit distinct banks

// One 16x16 f32 tile of  C = A(16xK) * W(nrow,:)^T  with K = 128, via 32
// chained V_WMMA_F32_16X16X4_F32.  ISA f32 A/B layout (16x4 / 4x16):
//   lanes 0-15 : VGPR0=K0, VGPR1=K1 ; lanes 16-31 : VGPR0=K2, VGPR1=K3
// so both A (from LDS) and B (= weight row, row-major W) use the same k index.
__device__ __forceinline__ v8f tile_gemm_f32(const float A[16][LDSP],
                                             const float* __restrict__ wrow,
                                             int lrow, int lhi) {
  v8f c = {};
#pragma unroll
  for (int k0 = 0; k0 < DD; k0 += 4) {
    const int k = k0 + 2 * lhi;
    v2f a; a.x = A[lrow][k]; a.y = A[lrow][k + 1];
    v2f b; b.x = wrow[k];    b.y = wrow[k + 1];
    c = __builtin_amdgcn_wmma_f32_16x16x4_f32(false, a, false, b,
                                              (short)0, c, false, false);
  }
  return c;
}

__device__ __forceinline__ float sigmoidf_(float v) {
  return 1.0f / (1.0f + __expf(-v));
}

// Block = 16 rows of B, 256 threads = 8 waves; wave w owns output columns
// [16w, 16w+16).  Chain: x -> elu(xW1^T+b1) -> xW2^T+b2 -> GRU(out, mem).
__global__ __launch_bounds__(256) void fused_mlp_gru(
    const float* __restrict__ x, const int* __restrict__ idx,
    const float* __restrict__ hist,
    const float* __restrict__ W1, const float* __restrict__ b1,
    const float* __restrict__ W2, const float* __restrict__ b2,
    const float* __restrict__ Wih, const float* __restrict__ bih,
    const float* __restrict__ Whh, const float* __restrict__ bhh,
    float* __restrict__ out, float* __restrict__ memout, int Nn) {
  __shared__ float xs[16][LDSP];
  __shared__ float h1s[16][LDSP];
  __shared__ float xfs[16][LDSP];
  __shared__ float mems[16][LDSP];
  __shared__ int   idxs[16];

  const int tid  = threadIdx.x;
  const int wave = tid >> 5;
  const int lane = tid & 31;
  const int lrow = lane & 15;   // N (for B/C frags) or M (for A frag)
  const int lhi  = lane >> 4;   // half-wave select
  const long long r0 = (long long)blockIdx.x * 16;

  if (tid < 16) idxs[tid] = idx[r0 + tid];
  __syncthreads();

  // Stage 0: load x tile + gather mem = history[last][idx]; emit mem output.
  for (int i = tid; i < 16 * DD; i += 256) {
    const int m = i >> 7, d = i & (DD - 1);
    xs[m][d] = x[(r0 + m) * DD + d];
    const float mv = hist[((long long)2 * Nn + idxs[m]) * DD + d];
    mems[m][d] = mv;
    memout[(r0 + m) * DD + d] = mv;
  }
  __syncthreads();

  // Stage 1: h1 = elu(x W1^T + b1).  C/D layout: lane n=lrow, m=8*lhi+v.
  {
    const int nb = wave * 16;
    v8f c = tile_gemm_f32(xs, W1 + (long long)(nb + lrow) * DD, lrow, lhi);
    const float bn = b1[nb + lrow];
#pragma unroll
    for (int v = 0; v < 8; ++v) {
      float t = c[v] + bn;
      t = (t > 0.0f) ? t : (__expf(t) - 1.0f);   // ELU, alpha=1
      h1s[8 * lhi + v][nb + lrow] = t;
    }
  }
  __syncthreads();

  // Stage 2: xf = h1 W2^T + b2.
  {
    const int nb = wave * 16;
    v8f c = tile_gemm_f32(h1s, W2 + (long long)(nb + lrow) * DD, lrow, lhi);
    const float bn = b2[nb + lrow];
#pragma unroll
    for (int v = 0; v < 8; ++v)
      xfs[8 * lhi + v][nb + lrow] = c[v] + bn;
  }
  __syncthreads();

  // Stage 3: gate pre-activations kept entirely in WMMA accumulators.
  // gi[t] = xf @ Wih[tD:(t+1)D]^T ; gh[t] = mem @ Whh[tD:(t+1)D]^T
  v8f cgi[3], cgh[3];
#pragma unroll
  for (int t = 0; t < 3; ++t) {
    const int nbg = t * DD + wave * 16;
    cgi[t] = tile_gemm_f32(xfs,  Wih + (long long)(nbg + lrow) * DD, lrow, lhi);
    cgh[t] = tile_gemm_f32(mems, Whh + (long long)(nbg + lrow) * DD, lrow, lhi);
  }

  // Stage 4: GRU gate math directly on fragments (lane n fixed, m = 8*lhi+v).
  const int   n    = wave * 16 + lrow;
  const float bir  = bih[n], biz = bih[DD + n], bin_ = bih[2 * DD + n];
  const float bhr  = bhh[n], bhz = bhh[DD + n], bhn  = bhh[2 * DD + n];
#pragma unroll
  for (int v = 0; v < 8; ++v) {
    const int   m  = 8 * lhi + v;
    const float r  = sigmoidf_(cgi[0][v] + bir + cgh[0][v] + bhr);
    const float z  = sigmoidf_(cgi[1][v] + biz + cgh[1][v] + bhz);
    const float nn = tanhf(cgi[2][v] + bin_ + r * (cgh[2][v] + bhn));
    const float h  = mems[m][n];
    out[(r0 + m) * DD + n] = (1.0f - z) * nn + z * h;
  }
}

// Bulk copy history -> new_history (1.5GB round trip; the roofline floor).
__global__ void copy_hist(const float4* __restrict__ src,
                          float4* __restrict__ dst, long long n4) {
  const long long i = (long long)blockIdx.x * blockDim.x + threadIdx.x;
  if (i < n4) dst[i] = src[i];
}

__global__ void init_claim(int* __restrict__ claim, int Nn) {
  const int i = blockIdx.x * blockDim.x + threadIdx.x;
  if (i < Nn) claim[i] = -1;
}

__global__ void claim_rows(const int* __restrict__ idx, int* __restrict__ claim,
                           int Bn) {
  const int b = blockIdx.x * blockDim.x + threadIdx.x;
  if (b < Bn) atomicMax(&claim[idx[b]], b);
}

// One wave per batch row: shift the deque + write out into the last slot.
// If a claim map is available, only the max-b duplicate writes (deterministic).
__global__ __launch_bounds__(256) void scatter_hist(
    const float* __restrict__ hist, const int* __restrict__ idx,
    const float* __restrict__ outv, const int* __restrict__ claim,
    float* __restrict__ nh, int Bn, int Nn) {
  const int b    = blockIdx.x * 8 + (threadIdx.x >> 5);
  const int lane = threadIdx.x & 31;
  if (b >= Bn) return;
  const long long row = idx[b];
  if (claim && claim[row] != b) return;

  const float4* s1 = (const float4*)(hist + ((long long)1 * Nn + row) * DD);
  const float4* s2 = (const float4*)(hist + ((long long)2 * Nn + row) * DD);
  const float4* so = (const float4*)(outv + (long long)b * DD);
  float4* d0 = (float4*)(nh + row * DD);
  float4* d1 = (float4*)(nh + ((long long)Nn + row) * DD);
  float4* d2 = (float4*)(nh + ((long long)2 * Nn + row) * DD);
  d0[lane] = s1[lane];   // slot0[idx] = old slot1[idx]
  d1[lane] = s2[lane];   // slot1[idx] = old slot2[idx]
  d2[lane] = so[lane];   // slot2[idx] = out[b]
}

extern "C" void kernel_launch(void* const* d_in, const int* in_sizes, int n_in,
                              void* d_out, int out_size, void* d_ws,
                              size_t ws_size, hipStream_t stream) {
  const float* x    = (const float*)d_in[0];
  const int*   idx  = (const int*)d_in[1];
  const float* hist = (const float*)d_in[2];
  const float* W1   = (const float*)d_in[3];
  const float* b1   = (const float*)d_in[4];
  const float* W2   = (const float*)d_in[5];
  const float* b2   = (const float*)d_in[6];
  const float* Wih  = (const float*)d_in[7];
  const float* bih  = (const float*)d_in[8];
  const float* Whh  = (const float*)d_in[9];
  const float* bhh  = (const float*)d_in[10];

  const int Bn = in_sizes[0] / DD;                           // 65536
  const long long hist_elems = (long long)in_sizes[2];       // 3*N*D
  const int Nn = (int)(hist_elems / (3LL * DD));             // 500000

  float* out    = (float*)d_out;
  float* memout = out + (long long)Bn * DD;
  float* nh     = memout + (long long)Bn * DD;

  // 1) fused MLP + GRU (WMMA) -> out, mem
  fused_mlp_gru<<<Bn / 16, 256, 0, stream>>>(x, idx, hist, W1, b1, W2, b2,
                                             Wih, bih, Whh, bhh, out, memout,
                                             Nn);

  // 2) bulk history copy
  const long long n4 = hist_elems / 4;
  copy_hist<<<(int)((n4 + 255) / 256), 256, 0, stream>>>(
      (const float4*)hist, (float4*)nh, n4);

  // 3) deterministic scatter of shifted rows (claim = max-b wins on dup idx)
  int* claim = nullptr;
  if (ws_size >= (size_t)Nn * sizeof(int)) {
    claim = (int*)d_ws;
    init_claim<<<(Nn + 255) / 256, 256, 0, stream>>>(claim, Nn);
    claim_rows<<<(Bn + 255) / 256, 256, 0, stream>>>(idx, claim, Bn);
  }
  scatter_hist<<<(Bn + 7) / 8, 256, 0, stream>>>(hist, idx, out, claim, nh,
                                                 Bn, Nn);
}